// PairWiseL2RefinementLoss_27805618274451
// MI455X (gfx1250) — compile-verified
//
#include <hip/hip_runtime.h>
#include <hip/hip_bf16.h>

#define BATCH 4
#define LSEQ 1024
#define NATOMS 3
#define NPTS (LSEQ * NATOMS)                  // 3072 points per batch
#define NTILES (NPTS / 16)                    // 192 tiles per axis
#define DIST_CLAMP_F 20.0f
#define ERR_CLAMP_F 25.0f
#define EPS_F 1e-5f

#define TPB 256
#define WAVES_PER_BLOCK (TPB / 32)

// Strip decomposition: each wave owns one (b, ti) row strip and a chunk of tj.
#define CHUNKS 16
#define TJ_PER_CHUNK (NTILES / CHUNKS)        // 12
#define STRIPS (BATCH * NTILES)               // 768
#define NWAVES (STRIPS * CHUNKS)              // 12288
#define BLOCKS2 (NWAVES / WAVES_PER_BLOCK)    // 1536

typedef __attribute__((ext_vector_type(2))) float v2f;
typedef __attribute__((ext_vector_type(4))) float v4f;
typedef __attribute__((ext_vector_type(8))) float v8f;

// ---------------------------------------------------------------------------
// Kernel 1: packed per-point records {x, y, z, |x|^2} for pred & gt, plain
// squared-norm arrays (for contiguous row-side v4f loads), replicated mask.
// ---------------------------------------------------------------------------
__global__ void pwl2_precompute(const float* __restrict__ gt,
                                const float* __restrict__ pred,
                                const float* __restrict__ mask,
                                v4f* __restrict__ cG,
                                v4f* __restrict__ cP,
                                float* __restrict__ sqnG,
                                float* __restrict__ sqnP,
                                float* __restrict__ mrep) {
    int n = blockIdx.x * blockDim.x + threadIdx.x;
    if (n >= BATCH * NPTS) return;
    const float* g = gt + (size_t)n * 3;
    const float* p = pred + (size_t)n * 3;
    float gx = g[0], gy = g[1], gz = g[2];
    float px = p[0], py = p[1], pz = p[2];
    float sg = gx * gx + gy * gy + gz * gz;
    float sp = px * px + py * py + pz * pz;
    cG[n] = (v4f){gx, gy, gz, sg};
    cP[n] = (v4f){px, py, pz, sp};
    sqnG[n] = sg;
    sqnP[n] = sp;
    int b = n / NPTS;
    int l = (n - b * NPTS) / NATOMS;
    mrep[n] = mask[b * LSEQ + l];
}

// ---------------------------------------------------------------------------
// Kernel 2: one wave per (strip, tj-chunk). Row-side operands hoisted out of
// the inner loop; inner loop per 16x16 tile: 3 vmem loads, 2 WMMAs, epilogue.
// A layout (ISA 7.12.2, 32-bit A 16x4): lanes 0-15 -> M=lane, VGPRs=K{0,1};
// lanes 16-31 -> K{2,3}. C/D: N = lane&15, M = r + (lane>=16 ? 8 : 0).
// ---------------------------------------------------------------------------
__global__ void __launch_bounds__(TPB)
pwl2_tiles(const v4f* __restrict__ cG,
           const v4f* __restrict__ cP,
           const float* __restrict__ sqnG,
           const float* __restrict__ sqnP,
           const float* __restrict__ mrep,
           float* __restrict__ partials) {
    __shared__ float wsum[WAVES_PER_BLOCK];

    const int lane = threadIdx.x & 31;
    const int wave = threadIdx.x >> 5;
    const int gwave = blockIdx.x * WAVES_PER_BLOCK + wave;  // < NWAVES
    const int half = lane >> 4;   // 0: K=0,1 (x,y)   1: K=2,3 (z,0)
    const int l15 = lane & 15;

    const int strip = gwave >> 4;            // (b, ti)
    const int chunk = gwave & (CHUNKS - 1);
    const int b = strip / NTILES;
    const int ti = strip - b * NTILES;
    const int tj0 = chunk * TJ_PER_CHUNK;

    const size_t baseN = (size_t)b * NPTS;
    const v4f* cGa = cG + baseN;
    const v4f* cPa = cP + baseN;
    const float* mrepa = mrep + baseN;

    // ---- Row-side setup (hoisted): A operands + row norms + row masks ----
    const int rowA = ti * 16 + l15;
    v4f rp = cPa[rowA];
    v4f rg = cGa[rowA];
    v2f aP = { half ? rp.z : rp.x, half ? 0.f : rp.y };
    v2f aG = { half ? rg.z : rg.x, half ? 0.f : rg.y };

    const size_t rbase = baseN + (size_t)ti * 16 + (size_t)half * 8;
    v4f sqPi0 = *(const v4f*)(sqnP + rbase);
    v4f sqPi1 = *(const v4f*)(sqnP + rbase + 4);
    v4f sqGi0 = *(const v4f*)(sqnG + rbase);
    v4f sqGi1 = *(const v4f*)(sqnG + rbase + 4);
    v4f mi0   = *(const v4f*)(mrep + rbase);
    v4f mi1   = *(const v4f*)(mrep + rbase + 4);

    float acc = 0.f;

    for (int tj = tj0; tj < tj0 + TJ_PER_CHUNK; ++tj) {
        const int colB = tj * 16 + l15;
        v4f cp = cPa[colB];          // {x, y, z, |x|^2} pred col
        v4f cg = cGa[colB];          // {x, y, z, |x|^2} gt col
        float mj = mrepa[colB];

        v2f bP = { half ? cp.z : cp.x, half ? 0.f : cp.y };
        v2f bG = { half ? cg.z : cg.x, half ? 0.f : cg.y };

        v8f czero = {};
        v8f gp = __builtin_amdgcn_wmma_f32_16x16x4_f32(
            false, aP, false, bP, (short)0, czero, false, false);
        v8f gg = __builtin_amdgcn_wmma_f32_16x16x4_f32(
            false, aG, false, bG, (short)0, czero, false, false);

        float tileSum = 0.f;
#pragma unroll
        for (int r = 0; r < 8; ++r) {
            float spi = (r < 4) ? sqPi0[r & 3] : sqPi1[r & 3];
            float sgi = (r < 4) ? sqGi0[r & 3] : sqGi1[r & 3];
            float mi  = (r < 4) ? mi0[r & 3]   : mi1[r & 3];

            // Raw HW sqrt (~1 ulp): inputs are clamped >= 0, magnitudes O(1e2),
            // no denormal-range concerns; loss is a tolerance-checked sum.
            float d2p = fmaxf(spi + cp.w - 2.0f * gp[r], 0.f);
            float dp  = fminf(__builtin_amdgcn_sqrtf(d2p), DIST_CLAMP_F);
            float d2g = fmaxf(sgi + cg.w - 2.0f * gg[r], 0.f);
            float dg  = __builtin_amdgcn_sqrtf(d2g);
            float diff = dp - dg;
            float e = fminf(diff * diff, ERR_CLAMP_F);
            tileSum = fmaf(e, mi, tileSum);
        }
        acc = fmaf(tileSum, mj, acc);
    }

    // Wave reduction (wave32).
#pragma unroll
    for (int off = 16; off > 0; off >>= 1)
        acc += __shfl_down(acc, off, 32);
    if (lane == 0) wsum[wave] = acc;
    __syncthreads();
    if (threadIdx.x == 0) {
        float s = 0.f;
#pragma unroll
        for (int w = 0; w < WAVES_PER_BLOCK; ++w) s += wsum[w];
        partials[blockIdx.x] = s;
    }
}

// ---------------------------------------------------------------------------
// Kernel 3: deterministic final reduction + analytic mask-sum + division.
// sum(M) = sum_b (NATOMS * sum_l mask[b,l])^2   (exact identity)
// ---------------------------------------------------------------------------
__global__ void pwl2_finalize(const float* __restrict__ partials,
                              const float* __restrict__ mask,
                              float* __restrict__ out) {
    __shared__ float red[TPB];
    float s = 0.f;
    for (int i = threadIdx.x; i < BLOCKS2; i += TPB) s += partials[i];
    red[threadIdx.x] = s;
    __syncthreads();
    for (int step = TPB / 2; step > 0; step >>= 1) {
        if ((int)threadIdx.x < step) red[threadIdx.x] += red[threadIdx.x + step];
        __syncthreads();
    }
    if (threadIdx.x == 0) {
        float sumM = 0.f;
        for (int bb = 0; bb < BATCH; ++bb) {
            float ms = 0.f;
            for (int l = 0; l < LSEQ; ++l) ms += mask[bb * LSEQ + l];
            ms *= (float)NATOMS;
            sumM += ms * ms;
        }
        out[0] = red[0] / (sumM + EPS_F);
    }
}

// ---------------------------------------------------------------------------
extern "C" void kernel_launch(void* const* d_in, const int* in_sizes, int n_in,
                              void* d_out, int out_size, void* d_ws, size_t ws_size,
                              hipStream_t stream) {
    const float* gt   = (const float*)d_in[0];  // (B, L, 3, 3) f32
    const float* pred = (const float*)d_in[1];  // (B, L, 3, 3) f32
    const float* mask = (const float*)d_in[2];  // (B, L) f32
    float* out = (float*)d_out;

    const int NTOT = BATCH * NPTS;              // 12288 points
    float* ws = (float*)d_ws;
    v4f* cG   = (v4f*)ws;                       // NTOT float4 (16B-aligned: ws is 256B-aligned)
    v4f* cP   = cG + NTOT;                      // NTOT float4
    float* fbase = (float*)(cP + NTOT);
    float* sqnG = fbase;                        // NTOT floats
    float* sqnP = sqnG + NTOT;                  // NTOT floats
    float* mrep = sqnP + NTOT;                  // NTOT floats
    float* partials = mrep + NTOT;              // BLOCKS2 floats

    pwl2_precompute<<<(NTOT + TPB - 1) / TPB, TPB, 0, stream>>>(
        gt, pred, mask, cG, cP, sqnG, sqnP, mrep);

    pwl2_tiles<<<BLOCKS2, TPB, 0, stream>>>(
        cG, cP, sqnG, sqnP, mrep, partials);

    pwl2_finalize<<<1, TPB, 0, stream>>>(partials, mask, out);
}